// NTM_86732569576124
// MI455X (gfx1250) — compile-verified
//
#include <hip/hip_runtime.h>
#include <hip/hip_bf16.h>
#include <math.h>

#define N_  128
#define H_  512
#define M_  2048
#define D_  128
#define G3H 1536
#define EPSV 1e-8f
#define LN2V 0.6931471805599453f

typedef __attribute__((ext_vector_type(2))) float v2f;
typedef __attribute__((ext_vector_type(4))) float v4f;
typedef __attribute__((ext_vector_type(8))) float v8f;

__device__ __forceinline__ float sigmoidf_(float x) { return 1.0f / (1.0f + __expf(-x)); }

// gfx1250 async copy: each lane copies 16 bytes from the global address src to the LDS address lds_off, tracked by ASYNCcnt.
__device__ __forceinline__ void async_b128(unsigned lds_off, const float* src) {
  asm volatile("global_load_async_to_lds_b128 %0, %1, off"
               :: "v"(lds_off), "v"(src) : "memory");
}

// ---------------------------------------------------------------------------
// K1: k = h_o_prev @ Wk^T + bk   (128x128, K=512) — fp32 WMMA, 1 wave / 16x16 tile
// ---------------------------------------------------------------------------
__global__ void k_gemm_k(const float* __restrict__ h, const float* __restrict__ Wk,
                         const float* __restrict__ bk, float* __restrict__ kout) {
  int wave = (blockIdx.x << 3) + (threadIdx.x >> 5);   // 0..63
  int lane = threadIdx.x & 31;
  int half = lane >> 4, l16 = lane & 15;
  int n0 = (wave >> 3) * 16, d0 = (wave & 7) * 16;
  v8f acc = {};
  const float* ap = h  + (size_t)(n0 + l16) * H_ + 2 * half;
  const float* bp = Wk + (size_t)(d0 + l16) * H_ + 2 * half;
  for (int kk = 0; kk < H_; kk += 4) {
    v2f a = *(const v2f*)(ap + kk);
    v2f b = *(const v2f*)(bp + kk);
    acc = __builtin_amdgcn_wmma_f32_16x16x4_f32(false, a, false, b, (short)0, acc, false, false);
  }
  int col = d0 + l16;
  float bc = bk[col];
  for (int v = 0; v < 8; ++v) {
    int row = n0 + v + 8 * half;
    kout[row * D_ + col] = acc[v] + bc;
  }
}

// ---------------------------------------------------------------------------
// K2: beta[n] = stable-softplus(h@Wb^T+bb)+(1-ln2); kb[n,d] = k[n,d]*beta/max(||k||,eps)
// ---------------------------------------------------------------------------
__global__ void k_beta_kb(const float* __restrict__ h, const float* __restrict__ Wb,
                          const float* __restrict__ bb, const float* __restrict__ kin,
                          float* __restrict__ kb) {
  int n = blockIdx.x, t = threadIdx.x;  // 128 threads
  __shared__ float red[128];
  float s = 0.f;
  for (int j = t; j < H_; j += 128) s += h[n * H_ + j] * Wb[j];
  red[t] = s; __syncthreads();
  for (int off = 64; off; off >>= 1) { if (t < off) red[t] += red[t + off]; __syncthreads(); }
  float x  = red[0] + bb[0];
  float bp = fmaxf(x, 0.f), bn = fminf(x, 0.f);
  float beta = log1pf(__expf(bn)) + bp + log1pf(__expf(-bp)) + (1.0f - LN2V);
  __syncthreads();
  float kv = kin[n * D_ + t];
  red[t] = kv * kv; __syncthreads();
  for (int off = 64; off; off >>= 1) { if (t < off) red[t] += red[t + off]; __syncthreads(); }
  float nk = fmaxf(sqrtf(red[0]), EPSV);
  kb[n * D_ + t] = kv * (beta / nk);
}

// ---------------------------------------------------------------------------
// K3: logits[n,m] = dot(c[n,m,:], kb[n,:]) / max(||c[n,m,:]||, eps)   (wave per row)
// ---------------------------------------------------------------------------
__global__ void k_logits(const float* __restrict__ c, const float* __restrict__ kb,
                         float* __restrict__ lout) {
  int row  = (blockIdx.x << 3) + (threadIdx.x >> 5);   // n*M + m
  int lane = threadIdx.x & 31;
  int n = row >> 11;                                   // M = 2048
  v4f cv = *((const v4f*)(c + (size_t)row * D_) + lane);
  v4f kv = *((const v4f*)(kb + n * D_) + lane);
  float dot = cv.x * kv.x + cv.y * kv.y + cv.z * kv.z + cv.w * kv.w;
  float ss  = cv.x * cv.x + cv.y * cv.y + cv.z * cv.z + cv.w * cv.w;
  for (int off = 16; off; off >>= 1) {
    dot += __shfl_xor(dot, off, 32);
    ss  += __shfl_xor(ss,  off, 32);
  }
  if (lane == 0) lout[row] = dot / fmaxf(sqrtf(ss), EPSV);
}

// ---------------------------------------------------------------------------
// K4: softmax over M per n (block per n, 256 thr x 8 elems)
// ---------------------------------------------------------------------------
__global__ void k_softmax(const float* __restrict__ lin, float* __restrict__ wout) {
  int n = blockIdx.x, t = threadIdx.x;
  __shared__ float red[256];
  float l[8], mx = -3.4e38f;
  for (int i = 0; i < 8; ++i) { l[i] = lin[n * M_ + t + (i << 8)]; mx = fmaxf(mx, l[i]); }
  red[t] = mx; __syncthreads();
  for (int off = 128; off; off >>= 1) { if (t < off) red[t] = fmaxf(red[t], red[t + off]); __syncthreads(); }
  mx = red[0]; __syncthreads();
  float s = 0.f;
  for (int i = 0; i < 8; ++i) { l[i] = __expf(l[i] - mx); s += l[i]; }
  red[t] = s; __syncthreads();
  for (int off = 128; off; off >>= 1) { if (t < off) red[t] += red[t + off]; __syncthreads(); }
  float inv = 1.0f / red[0];
  for (int i = 0; i < 8; ++i) wout[n * M_ + t + (i << 8)] = l[i] * inv;
}

// ---------------------------------------------------------------------------
// K5: partial r: rpart[p][n,:] = sum_{m in part p} w[n,m]*c_xf[n,m,:]
//     512 blocks x 16 waves; async global->LDS double-buffer (6 in flight, 8 slots)
// ---------------------------------------------------------------------------
__global__ void k_read_r(const float* __restrict__ c, const float* __restrict__ w,
                         float* __restrict__ rpart) {
  int n    = blockIdx.x >> 2;
  int part = blockIdx.x & 3;
  int wave = threadIdx.x >> 5, lane = threadIdx.x & 31;
  __shared__ float stage[16][8][128];     // 16 waves x 8 slots x 512B = 64KB
  __shared__ float acc_s[16 * 128];
  const int ROWS = 32;                    // 2048 / 4 parts / 16 waves
  int m0 = part * 512 + wave * ROWS;
  const float* cb = c + (size_t)n * M_ * D_;
  const float* wb = w + n * M_;
  unsigned lds0 = (unsigned)(size_t)(&stage[wave][0][0]);

  for (int i = 0; i < 6; ++i) {           // prologue: 6 rows in flight
    unsigned dst = lds0 + (unsigned)(i & 7) * 512u + (unsigned)lane * 16u;
    async_b128(dst, cb + (size_t)(m0 + i) * D_ + lane * 4);
  }
  v4f acc = {};
  for (int i = 0; i < ROWS - 6; ++i) {
    asm volatile("s_wait_asynccnt 0x5" ::: "memory");   // oldest row landed
    int slot = i & 7;
    float wm = wb[m0 + i];
    v4f cv = *(const v4f*)(&stage[wave][slot][lane * 4]);
    acc.x += wm * cv.x; acc.y += wm * cv.y; acc.z += wm * cv.z; acc.w += wm * cv.w;
    int j = i + 6;
    unsigned dst = lds0 + (unsigned)(j & 7) * 512u + (unsigned)lane * 16u;
    async_b128(dst, cb + (size_t)(m0 + j) * D_ + lane * 4);
  }
  asm volatile("s_wait_asynccnt 0x0" ::: "memory");     // drain
  for (int i = ROWS - 6; i < ROWS; ++i) {
    int slot = i & 7;
    float wm = wb[m0 + i];
    v4f cv = *(const v4f*)(&stage[wave][slot][lane * 4]);
    acc.x += wm * cv.x; acc.y += wm * cv.y; acc.z += wm * cv.z; acc.w += wm * cv.w;
  }
  int base = wave * 128 + lane * 4;
  acc_s[base] = acc.x; acc_s[base + 1] = acc.y; acc_s[base + 2] = acc.z; acc_s[base + 3] = acc.w;
  __syncthreads();
  if (threadIdx.x < 128) {
    float s = 0.f;
    for (int wv = 0; wv < 16; ++wv) s += acc_s[wv * 128 + threadIdx.x];
    rpart[(size_t)part * N_ * D_ + n * D_ + threadIdx.x] = s;
  }
}

// K5b: r = sum of 4 partials
__global__ void k_combine_r(const float* __restrict__ rpart, float* __restrict__ rout) {
  int i = blockIdx.x * 256 + threadIdx.x;               // N*D = 16384
  rout[i] = rpart[i] + rpart[N_ * D_ + i] + rpart[2 * N_ * D_ + i] + rpart[3 * N_ * D_ + i];
}

// ---------------------------------------------------------------------------
// K6a: gi = r@W_ih^T (K=128), gh = h@W_hh^T (K=512) — kept separate for GRU n-gate
// ---------------------------------------------------------------------------
__global__ void k_gru_gemm(const float* __restrict__ r, const float* __restrict__ h,
                           const float* __restrict__ Wih, const float* __restrict__ Whh,
                           float* __restrict__ gi, float* __restrict__ gh) {
  int wave = (blockIdx.x << 3) + (threadIdx.x >> 5);   // 0..767
  int lane = threadIdx.x & 31, half = lane >> 4, l16 = lane & 15;
  int n0 = (wave & 7) * 16, j0 = (wave >> 3) * 16;
  v8f ai = {}, ah = {};
  {
    const float* ap = r   + (size_t)(n0 + l16) * D_ + 2 * half;
    const float* bp = Wih + (size_t)(j0 + l16) * D_ + 2 * half;
    for (int kk = 0; kk < D_; kk += 4) {
      v2f a = *(const v2f*)(ap + kk);
      v2f b = *(const v2f*)(bp + kk);
      ai = __builtin_amdgcn_wmma_f32_16x16x4_f32(false, a, false, b, (short)0, ai, false, false);
    }
  }
  {
    const float* ap = h   + (size_t)(n0 + l16) * H_ + 2 * half;
    const float* bp = Whh + (size_t)(j0 + l16) * H_ + 2 * half;
    for (int kk = 0; kk < H_; kk += 4) {
      v2f a = *(const v2f*)(ap + kk);
      v2f b = *(const v2f*)(bp + kk);
      ah = __builtin_amdgcn_wmma_f32_16x16x4_f32(false, a, false, b, (short)0, ah, false, false);
    }
  }
  int col = j0 + l16;
  for (int v = 0; v < 8; ++v) {
    int row = n0 + v + 8 * half;
    gi[row * G3H + col] = ai[v];
    gh[row * G3H + col] = ah[v];
  }
}

// ---------------------------------------------------------------------------
// K6b: GRU gating -> h_o (written directly into d_out)
// ---------------------------------------------------------------------------
__global__ void k_gru_gate(const float* __restrict__ gi, const float* __restrict__ gh,
                           const float* __restrict__ bih, const float* __restrict__ bhh,
                           const float* __restrict__ hprev, float* __restrict__ hout) {
  int idx = blockIdx.x * 256 + threadIdx.x;            // N*H
  int n = idx >> 9, j = idx & 511;
  int b = n * G3H + j;
  float ir = gi[b]          + bih[j],          hr = gh[b]          + bhh[j];
  float iz = gi[b + H_]     + bih[H_ + j],     hz = gh[b + H_]     + bhh[H_ + j];
  float in_= gi[b + 2 * H_] + bih[2 * H_ + j], hn = gh[b + 2 * H_] + bhh[2 * H_ + j];
  float rg = sigmoidf_(ir + hr);
  float z  = sigmoidf_(iz + hz);
  float ng = tanhf(in_ + rg * hn);
  hout[idx] = (1.0f - z) * ng + z * hprev[idx];
}

// ---------------------------------------------------------------------------
// K7: e = sigmoid(h_o@We^T+be), v = h_o@Wv^T+bv   (128 waves: 64 for e, 64 for v)
// ---------------------------------------------------------------------------
__global__ void k_ev_gemm(const float* __restrict__ ho, const float* __restrict__ We,
                          const float* __restrict__ be, const float* __restrict__ Wv,
                          const float* __restrict__ bv, float* __restrict__ e,
                          float* __restrict__ v) {
  int wave = (blockIdx.x << 3) + (threadIdx.x >> 5);   // 0..127
  int lane = threadIdx.x & 31, half = lane >> 4, l16 = lane & 15;
  int sel = wave >> 6, t = wave & 63;
  int n0 = (t >> 3) * 16, d0 = (t & 7) * 16;
  const float* W = sel ? Wv : We;
  v8f acc = {};
  const float* ap = ho + (size_t)(n0 + l16) * H_ + 2 * half;
  const float* bp = W  + (size_t)(d0 + l16) * H_ + 2 * half;
  for (int kk = 0; kk < H_; kk += 4) {
    v2f a = *(const v2f*)(ap + kk);
    v2f b = *(const v2f*)(bp + kk);
    acc = __builtin_amdgcn_wmma_f32_16x16x4_f32(false, a, false, b, (short)0, acc, false, false);
  }
  int col = d0 + l16;
  float bias = sel ? bv[col] : be[col];
  float* out = sel ? v : e;
  for (int vv = 0; vv < 8; ++vv) {
    int row = n0 + vv + 8 * half;
    float val = acc[vv] + bias;
    if (!sel) val = sigmoidf_(val);
    out[row * D_ + col] = val;
  }
}

// ---------------------------------------------------------------------------
// K8: c_new = c_prev*(1 - w_kc*e) + w_kc*v   (float4 streaming, NT load/store)
// ---------------------------------------------------------------------------
__global__ void k_update(const float* __restrict__ cprev, const float* __restrict__ wkc,
                         const float* __restrict__ e, const float* __restrict__ v,
                         float* __restrict__ cnew) {
  int idx = blockIdx.x * 256 + threadIdx.x;            // over N*M*D/4
  int d4  = idx & 31;
  int row = idx >> 5;
  int n   = row >> 11;
  float w = wkc[row];
  v4f ev = *((const v4f*)(e + n * D_) + d4);
  v4f vv = *((const v4f*)(v + n * D_) + d4);
  v4f cv = __builtin_nontemporal_load((const v4f*)cprev + idx);
  v4f o;
  o.x = cv.x * (1.f - w * ev.x) + w * vv.x;
  o.y = cv.y * (1.f - w * ev.y) + w * vv.y;
  o.z = cv.z * (1.f - w * ev.z) + w * vv.z;
  o.w = cv.w * (1.f - w * ev.w) + w * vv.w;
  __builtin_nontemporal_store(o, (v4f*)cnew + idx);
}

// ---------------------------------------------------------------------------
extern "C" void kernel_launch(void* const* d_in, const int* in_sizes, int n_in,
                              void* d_out, int out_size, void* d_ws, size_t ws_size,
                              hipStream_t stream) {
  (void)in_sizes; (void)n_in; (void)out_size; (void)ws_size;
  const float* h     = (const float*)d_in[0];
  const float* cprev = (const float*)d_in[1];
  const float* cxf   = (const float*)d_in[2];
  const float* Wk    = (const float*)d_in[3];
  const float* bk    = (const float*)d_in[4];
  const float* Wb    = (const float*)d_in[5];
  const float* bb    = (const float*)d_in[6];
  const float* We    = (const float*)d_in[7];
  const float* be    = (const float*)d_in[8];
  const float* Wv    = (const float*)d_in[9];
  const float* bv    = (const float*)d_in[10];
  const float* Wih   = (const float*)d_in[11];
  const float* Whh   = (const float*)d_in[12];
  const float* bih   = (const float*)d_in[13];
  const float* bhh   = (const float*)d_in[14];

  float* out  = (float*)d_out;
  float* hout = out;              // h_o: N*H floats
  float* cnew = out + N_ * H_;    // c_new: N*M*D floats

  float* ws    = (float*)d_ws;
  float* kbuf  = ws; ws += N_ * D_;
  float* kb    = ws; ws += N_ * D_;
  float* logit = ws; ws += N_ * M_;
  float* wkx   = ws; ws += N_ * M_;
  float* wkc   = ws; ws += N_ * M_;
  float* gi    = ws; ws += N_ * G3H;
  float* gh    = ws; ws += N_ * G3H;
  float* rpart = ws; ws += 4 * N_ * D_;
  float* rbuf  = ws; ws += N_ * D_;
  float* ebuf  = ws; ws += N_ * D_;
  float* vbuf  = ws; ws += N_ * D_;

  // Ordered for L2 residency: c_xf passes back-to-back, then c_prev passes.
  k_gemm_k   <<<8,   256, 0, stream>>>(h, Wk, bk, kbuf);
  k_beta_kb  <<<N_,  128, 0, stream>>>(h, Wb, bb, kbuf, kb);
  k_logits   <<<(N_*M_)/8, 256, 0, stream>>>(cxf, kb, logit);
  k_softmax  <<<N_,  256, 0, stream>>>(logit, wkx);
  k_read_r   <<<N_*4, 512, 0, stream>>>(cxf, wkx, rpart);
  k_combine_r<<<(N_*D_)/256, 256, 0, stream>>>(rpart, rbuf);
  k_gru_gemm <<<96,  256, 0, stream>>>(rbuf, h, Wih, Whh, gi, gh);
  k_gru_gate <<<(N_*H_)/256, 256, 0, stream>>>(gi, gh, bih, bhh, h, hout);
  k_ev_gemm  <<<16,  256, 0, stream>>>(hout, We, be, Wv, bv, ebuf, vbuf);
  k_logits   <<<(N_*M_)/8, 256, 0, stream>>>(cprev, kb, logit);
  k_softmax  <<<N_,  256, 0, stream>>>(logit, wkc);
  k_update   <<<(N_*M_*(D_/4))/256, 256, 0, stream>>>(cprev, wkc, ebuf, vbuf, cnew);
}